// SAGKT_49022756717178
// MI455X (gfx1250) — compile-verified
//
#include <hip/hip_runtime.h>
#include <hip/hip_bf16.h>
#include <math.h>

// ---------------------------------------------------------------------------
// SAGKT EGAT forward for MI455X (gfx1250, wave32).
// Memory-bound graph scatter/gather; dense projections use fp32 WMMA
// (V_WMMA_F32_16X16X4_F32) with the A-tile staged into LDS once per
// workgroup by the Tensor Data Mover (tensor_load_to_lds / TENSORcnt).
// All gather targets are L2-resident (192MB L2).
// ---------------------------------------------------------------------------

#define N_NODES 50000
#define N_EDGES 800000
#define BATCH   1024

typedef float v2f __attribute__((ext_vector_type(2)));
typedef float v8f __attribute__((ext_vector_type(8)));
typedef unsigned int u32x4 __attribute__((ext_vector_type(4)));
typedef int i32x4 __attribute__((ext_vector_type(4)));
typedef int i32x8 __attribute__((ext_vector_type(8)));

// ---------------- atomic float max (signed/unsigned int ordering trick) ----
__device__ __forceinline__ void atomicMaxF(float* addr, float val) {
    if (val >= 0.0f) {
        atomicMax((int*)addr, __float_as_int(val));
    } else {
        atomicMin((unsigned int*)addr, __float_as_uint(val));
    }
}

// ---------------- init: zero agg/den, m = -inf -----------------------------
__global__ void k_init(float* __restrict__ agg, float* __restrict__ m,
                       float* __restrict__ den, int n128, int n4) {
    int i = blockIdx.x * blockDim.x + threadIdx.x;
    if (i < n128) agg[i] = 0.0f;
    if (i < n4) { den[i] = 0.0f; m[i] = -__builtin_inff(); }
}

// ---------------- node projections via fp32 WMMA + TDM staging -------------
// Computes ni = X@Wni.T [N,32], nj = X@Wnj.T [N,32], ft = X@Wnode.T+b [N,128]
// One 384-thread block (12 waves) per 16-row node tile. Wave w computes
// output column tile w: ot 0-1 -> ni, ot 2-3 -> nj, ot 4-11 -> ft.
// The 16x32 fp32 A-tile is DMA'd into LDS once (TDM), then each wave builds
// A fragments from LDS; K = 32 handled as 8 x (16x16x4) WMMA steps.
__global__ __launch_bounds__(384)
void k_nodeproj(const float* __restrict__ X, int ldx,
                const float* __restrict__ Wni, const float* __restrict__ Wnj,
                const float* __restrict__ Wnode, const float* __restrict__ Bnode,
                float* __restrict__ ni, float* __restrict__ nj,
                float* __restrict__ ft) {
    __shared__ float sX[16 * 32];               // one 16x32 fp32 node tile (2KB)
    const int tile = blockIdx.x;
    const int wid  = threadIdx.x >> 5;          // 0..11: output column tile
    const int lane = threadIdx.x & 31;

#if __has_builtin(__builtin_amdgcn_tensor_load_to_lds)
    // ---- stage A-tile into LDS with the Tensor Data Mover (wave 0 only) ---
    if (wid == 0) {                             // wave-uniform; TDM ignores EXEC
        unsigned long long ga =
            (unsigned long long)(const void*)(X + (size_t)tile * 16 * (size_t)ldx);
        unsigned int la = (unsigned int)(unsigned long long)(const void*)&sX[0];
        // D# group 0: count=1 | lds_addr | global_addr[56:0] | type=2
        u32x4 g0 = { 1u,
                     la,
                     (unsigned int)ga,
                     ((unsigned int)((ga >> 32) & 0x01FFFFFFull)) | (2u << 30) };
        // D# group 1: data_size=4B(code 2), tensor 32 x 16, tile 32 x 16,
        //             tensor_dim0_stride = ldx (data_size units)
        i32x8 g1 = { (int)(2u << 16),           // [17:16] data_size = 2 (4B)
                     (int)(32u << 16),          // [63:48] tensor_dim0 lo16 = 32
                     (int)(16u << 16),          // [95:80] tensor_dim1 lo16 = 16
                     (int)(32u << 16),          // [127:112] tile_dim0 = 32
                     16,                        // [143:128] tile_dim1 = 16
                     (int)(unsigned)ldx,        // [191:160] tensor_dim0_stride lo32
                     0, 0 };
        i32x4 gz = { 0, 0, 0, 0 };              // 2-D tile: groups 2/3 unused
#if defined(__clang_major__) && (__clang_major__ >= 23)
        i32x8 gz8 = { 0, 0, 0, 0, 0, 0, 0, 0 };
        __builtin_amdgcn_tensor_load_to_lds(g0, g1, gz, gz, gz8, 0);
#else
        __builtin_amdgcn_tensor_load_to_lds(g0, g1, gz, gz, 0);
#endif
        __builtin_amdgcn_s_wait_tensorcnt(0);
    }
    __syncthreads();
#else
    // fallback: cooperative load of the A-tile
    for (int i = threadIdx.x; i < 16 * 32; i += (int)blockDim.x) {
        int r = i >> 5, c = i & 31;
        sX[i] = X[((size_t)tile * 16 + r) * (size_t)ldx + c];
    }
    __syncthreads();
#endif

    const float* W; float* D; int ldd; int colbase; const float* bias = nullptr;
    if (wid < 2)      { W = Wni;   D = ni; ldd = 32;  colbase = wid * 16; }
    else if (wid < 4) { W = Wnj;   D = nj; ldd = 32;  colbase = (wid - 2) * 16; }
    else              { W = Wnode; D = ft; ldd = 128; colbase = (wid - 4) * 16; bias = Bnode; }

    int half = lane >> 4;                       // 0: lanes 0-15, 1: lanes 16-31
    int m16  = lane & 15;
    int col  = colbase + m16;                   // B/C column (N)
    const float* Wrow = W + col * 32;           // weight row [out,in], in=32

    v8f c = {};                                 // fp32 accumulator, 16x16 tile
#pragma unroll
    for (int kk = 0; kk < 8; ++kk) {
        int kb = kk * 4 + half * 2;             // ISA layout: half selects K pair
        v2f a, b;
        a.x = sX[m16 * 32 + kb];                // A fragment from LDS (ds_load)
        a.y = sX[m16 * 32 + kb + 1];
        b.x = Wrow[kb];
        b.y = Wrow[kb + 1];
        c = __builtin_amdgcn_wmma_f32_16x16x4_f32(
                /*neg_a=*/false, a, /*neg_b=*/false, b,
                /*c_mod=*/(short)0, c, /*reuse_a=*/false, /*reuse_b=*/false);
    }

    float bv = bias ? bias[col] : 0.0f;
#pragma unroll
    for (int r = 0; r < 8; ++r) {               // C layout: VGPR r -> M = r + 8*half
        int mr = tile * 16 + r + 8 * half;
        D[mr * ldd + col] = c[r] + bv;
    }
}

// ---------------- edge phase 1: f, logits, segment max ---------------------
__global__ __launch_bounds__(256)
void k_edge_logits(const float* __restrict__ efeat,
                   const int* __restrict__ src, const int* __restrict__ dst,
                   const float* __restrict__ ni, const float* __restrict__ nj,
                   const float* __restrict__ wfij, const float* __restrict__ be,
                   const float* __restrict__ attn,
                   float* __restrict__ lg, float* __restrict__ m, int E) {
    __shared__ float sW[32 * 8];
    __shared__ float sB[32];
    __shared__ float sA[32];
    int t = threadIdx.x;
    sW[t] = wfij[t];                            // blockDim == 256 == 32*8
    if (t < 32) { sB[t] = be[t]; sA[t] = attn[t]; }
    __syncthreads();

    int e = blockIdx.x * 256 + t;
    if (e >= E) return;

    float ef[8];
#pragma unroll
    for (int j = 0; j < 8; ++j) ef[j] = efeat[e * 8 + j];
    int s = src[e], d = dst[e];
    const float* nis = ni + s * 32;
    const float* njd = nj + d * 32;

#pragma unroll
    for (int h = 0; h < 4; ++h) {
        float logit = 0.0f;
#pragma unroll
        for (int j = 0; j < 8; ++j) {
            int o = h * 8 + j;
            float v = sB[o] + nis[o] + njd[o];
#pragma unroll
            for (int k = 0; k < 8; ++k) v += sW[o * 8 + k] * ef[k];
            v = v > 0.0f ? v : 0.01f * v;       // LeakyReLU(0.01)
            logit += v * sA[o];
        }
        lg[e * 4 + h] = logit;
        atomicMaxF(m + d * 4 + h, logit);
    }
}

// ---------------- edge phase 2: exp + segment sum --------------------------
__global__ __launch_bounds__(256)
void k_edge_exp(const int* __restrict__ dst, const float* __restrict__ m,
                float* __restrict__ lg, float* __restrict__ den, int E) {
    int e = blockIdx.x * blockDim.x + threadIdx.x;
    if (e >= E) return;
    int d = dst[e];
#pragma unroll
    for (int h = 0; h < 4; ++h) {
        float ex = expf(lg[e * 4 + h] - m[d * 4 + h]);
        lg[e * 4 + h] = ex;                     // reuse buffer for exp values
        atomicAdd(den + d * 4 + h, ex);
    }
}

// ---------------- edge phase 3: normalize + weighted scatter ---------------
// 128 threads per edge (H=4 heads x Fn=32): coalesced gather of ft[src],
// coalesced atomicAdd scatter into agg[dst]. All arrays L2-resident.
__global__ __launch_bounds__(256)
void k_edge_agg(const int* __restrict__ src, const int* __restrict__ dst,
                const float* __restrict__ ft, const float* __restrict__ lg,
                const float* __restrict__ den, float* __restrict__ agg, int total) {
    int i = blockIdx.x * blockDim.x + threadIdx.x;
    if (i >= total) return;
    int e = i >> 7;
    int t = i & 127;
    int h = t >> 5;
    int s = src[e], d = dst[e];
    float a = lg[e * 4 + h] / den[d * 4 + h];
    atomicAdd(agg + d * 128 + t, ft[s * 128 + t] * a);
}

// ---------------- node finish: sum heads + ELU -> states column ------------
__global__ __launch_bounds__(256)
void k_node_finish(const float* __restrict__ agg, float* __restrict__ states,
                   int layer, int N) {
    int i = blockIdx.x * blockDim.x + threadIdx.x;
    if (i >= N * 32) return;
    int n = i >> 5, c = i & 31;
    const float* r = agg + n * 128 + c;
    float s = r[0] + r[32] + r[64] + r[96];
    float v = s > 0.0f ? s : expm1f(s);         // ELU(alpha=1)
    states[n * 64 + layer * 32 + c] = v;
}

// ---------------- MLP head -------------------------------------------------
__global__ __launch_bounds__(256)
void k_mlp1(const float* __restrict__ states, const int* __restrict__ uidx,
            const int* __restrict__ iidx, const float* __restrict__ w1,
            const float* __restrict__ b1, float* __restrict__ hid) {
    int i = blockIdx.x * blockDim.x + threadIdx.x;
    if (i >= BATCH * 128) return;
    int b = i >> 7, o = i & 127;
    const float* zu = states + uidx[b] * 64;
    const float* zi = states + iidx[b] * 64;
    const float* w = w1 + o * 128;
    float acc = b1[o];
#pragma unroll 8
    for (int k = 0; k < 64; ++k) acc += zu[k] * w[k];
#pragma unroll 8
    for (int k = 0; k < 64; ++k) acc += zi[k] * w[k + 64];
    hid[i] = acc > 0.0f ? acc : 0.0f;
}

__global__ __launch_bounds__(256)
void k_mlp2(const float* __restrict__ hid, const float* __restrict__ w2,
            const float* __restrict__ b2, float* __restrict__ out) {
    int b = blockIdx.x * blockDim.x + threadIdx.x;
    if (b >= BATCH) return;
    const float* h = hid + b * 128;
    float acc = b2[0];
#pragma unroll 8
    for (int k = 0; k < 128; ++k) acc += h[k] * w2[k];
    out[b] = 1.0f / (1.0f + expf(-acc));
}

// ---------------------------------------------------------------------------
static inline size_t align256(size_t x) { return (x + 255) & ~(size_t)255; }

extern "C" void kernel_launch(void* const* d_in, const int* in_sizes, int n_in,
                              void* d_out, int out_size, void* d_ws, size_t ws_size,
                              hipStream_t stream) {
    (void)in_sizes; (void)n_in; (void)out_size; (void)ws_size;

    const float* x      = (const float*)d_in[0];
    const float* efeat  = (const float*)d_in[1];
    const float* w_ni[2]   = {(const float*)d_in[2],  (const float*)d_in[9]};
    const float* w_nj[2]   = {(const float*)d_in[3],  (const float*)d_in[10]};
    const float* w_fij[2]  = {(const float*)d_in[4],  (const float*)d_in[11]};
    const float* b_e[2]    = {(const float*)d_in[5],  (const float*)d_in[12]};
    const float* attn[2]   = {(const float*)d_in[6],  (const float*)d_in[13]};
    const float* w_node[2] = {(const float*)d_in[7],  (const float*)d_in[14]};
    const float* b_node[2] = {(const float*)d_in[8],  (const float*)d_in[15]};
    const float* lin1_w = (const float*)d_in[16];
    const float* lin1_b = (const float*)d_in[17];
    const float* lin2_w = (const float*)d_in[18];
    const float* lin2_b = (const float*)d_in[19];
    const int*   src    = (const int*)d_in[20];
    const int*   dst    = (const int*)d_in[21];
    const int*   uidx   = (const int*)d_in[22];
    const int*   iidx   = (const int*)d_in[23];
    float* out = (float*)d_out;

    // workspace carve-out
    char* ws = (char*)d_ws;
    float* ni  = (float*)ws; ws += align256((size_t)N_NODES * 32 * 4);
    float* nj  = (float*)ws; ws += align256((size_t)N_NODES * 32 * 4);
    float* ft  = (float*)ws; ws += align256((size_t)N_NODES * 128 * 4);
    float* lg  = (float*)ws; ws += align256((size_t)N_EDGES * 4 * 4);
    float* m   = (float*)ws; ws += align256((size_t)N_NODES * 4 * 4);
    float* den = (float*)ws; ws += align256((size_t)N_NODES * 4 * 4);
    float* agg = (float*)ws; ws += align256((size_t)N_NODES * 128 * 4);
    float* states = (float*)ws; ws += align256((size_t)N_NODES * 64 * 4);
    float* hid = (float*)ws; ws += align256((size_t)BATCH * 128 * 4);

    const int ntiles = N_NODES / 16;            // 3125, exact
    const int n128 = N_NODES * 128;
    const int n4 = N_NODES * 4;
    const int aggTotal = N_EDGES * 128;

    for (int layer = 0; layer < 2; ++layer) {
        const float* X = (layer == 0) ? x : states;
        int ldx = (layer == 0) ? 32 : 64;       // layer-1 input = states[:, :32]

        k_init<<<(n128 + 255) / 256, 256, 0, stream>>>(agg, m, den, n128, n4);
        k_nodeproj<<<ntiles, 384, 0, stream>>>(X, ldx, w_ni[layer], w_nj[layer],
                                               w_node[layer], b_node[layer],
                                               ni, nj, ft);
        k_edge_logits<<<(N_EDGES + 255) / 256, 256, 0, stream>>>(
            efeat, src, dst, ni, nj, w_fij[layer], b_e[layer], attn[layer],
            lg, m, N_EDGES);
        k_edge_exp<<<(N_EDGES + 255) / 256, 256, 0, stream>>>(dst, m, lg, den, N_EDGES);
        k_edge_agg<<<(aggTotal + 255) / 256, 256, 0, stream>>>(src, dst, ft, lg, den,
                                                               agg, aggTotal);
        k_node_finish<<<(N_NODES * 32 + 255) / 256, 256, 0, stream>>>(agg, states,
                                                                      layer, N_NODES);
    }

    k_mlp1<<<(BATCH * 128 + 255) / 256, 256, 0, stream>>>(states, uidx, iidx,
                                                          lin1_w, lin1_b, hid);
    k_mlp2<<<(BATCH + 255) / 256, 256, 0, stream>>>(hid, lin2_w, lin2_b, out);
}